// CPProfileSurrogate_11046655885317
// MI455X (gfx1250) — compile-verified
//
#include <hip/hip_runtime.h>
#include <hip/hip_bf16.h>
#include <math.h>

typedef __attribute__((ext_vector_type(2))) float v2f;
typedef __attribute__((ext_vector_type(8))) float v8f;

#define B_SZ   32768
#define S_MC   4
#define D_IN   256
#define LAT    8
#define NBINS  300
#define KPIX   1323
#define RANK   4

// ---------------------------------------------------------------------------
// Phase 1: encoder GEMM via V_WMMA_F32_16X16X4_F32.
// Weight matrix packed 256x16: cols 0..7 = mu_w rows, cols 8..15 = std_w rows,
// so one 16x16 WMMA tile yields mu_pre and std_pre for 16 reflections at once.
// Epilogue fuses bias + softplus + rsample + (h@C) + A[g] gather, emitting the
// per-row rank-4 coefficient vectors consumed by phase 2.
// Block = 256 threads = 8 waves, each wave owns one 16-row tile (128 rows/blk).
// ---------------------------------------------------------------------------
__global__ __launch_bounds__(256)
void phase1_encoder(const float* __restrict__ x,
                    const float* __restrict__ eps,
                    const float* __restrict__ A,
                    const float* __restrict__ C,
                    const float* __restrict__ mu_w,
                    const float* __restrict__ mu_b,
                    const float* __restrict__ std_w,
                    const float* __restrict__ std_b,
                    const int*   __restrict__ group_labels,
                    float* __restrict__ out_mu,    // (B, 8)
                    float* __restrict__ out_std,   // (B, 8)
                    float* __restrict__ w_vecs)    // (B, 5, 4)
{
    __shared__ float Wlds[D_IN * 16];     // packed weights [k][n]
    __shared__ float Dtile[8][16 * 16];   // per-wave 16x16 result
    __shared__ float Clds[LAT * RANK];

    const int tid = threadIdx.x;

    for (int i = tid; i < D_IN * 16; i += 256) {
        const int n = i & 15, k = i >> 4;
        Wlds[i] = (n < 8) ? mu_w[n * D_IN + k] : std_w[(n - 8) * D_IN + k];
    }
    if (tid < LAT * RANK) Clds[tid] = C[tid];
    __syncthreads();

    const int wave = tid >> 5;
    const int lane = tid & 31;
    const int half = lane >> 4;     // K-half selector per ISA A/B layout
    const int l    = lane & 15;
    const int row0 = (blockIdx.x * 8 + wave) * 16;

    v8f acc = {};
    const float* xrow = x + (size_t)(row0 + l) * D_IN + 2 * half;
    #pragma unroll 4
    for (int k0 = 0; k0 < D_IN; k0 += 4) {
        v2f a, bm;
        a.x  = xrow[k0];
        a.y  = xrow[k0 + 1];
        bm.x = Wlds[(k0 + 2 * half) * 16 + l];
        bm.y = Wlds[(k0 + 2 * half + 1) * 16 + l];
        acc = __builtin_amdgcn_wmma_f32_16x16x4_f32(false, a, false, bm,
                                                    (short)0, acc, false, false);
    }
    #pragma unroll
    for (int r = 0; r < 8; ++r)
        Dtile[wave][(r + 8 * half) * 16 + l] = acc[r];   // D[M=r+8*half][N=l]
    __syncthreads();

    if (tid < 128) {
        const int wt  = tid >> 4, rr = tid & 15;
        const int row = blockIdx.x * 128 + tid;

        float mu[LAT], sd[LAT];
        #pragma unroll
        for (int n = 0; n < LAT; ++n) {
            mu[n] = Dtile[wt][rr * 16 + n] + mu_b[n];
            const float sp = Dtile[wt][rr * 16 + 8 + n] + std_b[n];
            sd[n] = (sp > 0.f) ? (sp + log1pf(__expf(-sp))) : log1pf(__expf(sp));
            out_mu[row * LAT + n]  = mu[n];
            out_std[row * LAT + n] = sd[n];
        }

        const int g = group_labels[row];
        const float4 An = *(const float4*)(A + (size_t)g * RANK);

        {   // mean row (index 4)
            float z[RANK] = {0.f, 0.f, 0.f, 0.f};
            #pragma unroll
            for (int n = 0; n < LAT; ++n)
                #pragma unroll
                for (int r2 = 0; r2 < RANK; ++r2)
                    z[r2] = fmaf(mu[n], Clds[n * RANK + r2], z[r2]);
            *(float4*)(w_vecs + ((size_t)row * 5 + 4) * RANK) =
                make_float4(An.x * z[0], An.y * z[1], An.z * z[2], An.w * z[3]);
        }
        #pragma unroll
        for (int s = 0; s < S_MC; ++s) {
            const float* ep = eps + ((size_t)s * B_SZ + row) * LAT;
            float z[RANK] = {0.f, 0.f, 0.f, 0.f};
            #pragma unroll
            for (int n = 0; n < LAT; ++n) {
                const float h = fmaf(sd[n], ep[n], mu[n]);
                #pragma unroll
                for (int r2 = 0; r2 < RANK; ++r2)
                    z[r2] = fmaf(h, Clds[n * RANK + r2], z[r2]);
            }
            *(float4*)(w_vecs + ((size_t)row * 5 + s) * RANK) =
                make_float4(An.x * z[0], An.y * z[1], An.z * z[2], An.w * z[3]);
        }
    }
}

// ---------------------------------------------------------------------------
// Phase 2: streaming logits + softmax. One block per (sample|mean, reflection).
// Bandwidth-bound: writes each of the 868 MB of output exactly once.
// ---------------------------------------------------------------------------
#define P2_THREADS 256
#define P2_ITERS   6   // ceil(1323/256)

__device__ __forceinline__ float waveMax(float v) {
    #pragma unroll
    for (int o = 16; o > 0; o >>= 1) v = fmaxf(v, __shfl_xor(v, o, 32));
    return v;
}
__device__ __forceinline__ float waveSum(float v) {
    #pragma unroll
    for (int o = 16; o > 0; o >>= 1) v += __shfl_xor(v, o, 32);
    return v;
}

__global__ __launch_bounds__(P2_THREADS)
void phase2_softmax(const float* __restrict__ Bmat,    // (K, 4)
                    const float* __restrict__ bias,    // (NBINS, K)
                    const int*   __restrict__ group_labels,
                    const float* __restrict__ w_vecs,  // (B, 5, 4)
                    float* __restrict__ zp,            // (4, B, K)
                    float* __restrict__ mean_profile)  // (B, K)
{
    __shared__ float4 Blds[KPIX];
    __shared__ float  red[8];

    const unsigned tid = threadIdx.x;
    const unsigned rid = blockIdx.x;
    const unsigned b   = rid & (B_SZ - 1);
    const unsigned s   = rid >> 15;          // 0..3 samples, 4 = mean

    const int g = group_labels[b];
    const float* brow = bias + (size_t)g * KPIX;
    __builtin_prefetch(brow, 0, 1);          // global_prefetch_b8

    for (int k = tid; k < KPIX; k += P2_THREADS)
        Blds[k] = *(const float4*)(Bmat + (size_t)k * RANK);

    const float4 w = *(const float4*)(w_vecs + ((size_t)b * 5 + s) * RANK);
    __syncthreads();

    const int wave = tid >> 5, lane = tid & 31;

    float ex[P2_ITERS];
    float lmax = -3.402823466e38f;
    #pragma unroll
    for (int i = 0; i < P2_ITERS; ++i) {
        const int k = tid + i * P2_THREADS;
        float v = -3.402823466e38f;
        if (k < KPIX) {
            const float4 Bk = Blds[k];
            v = fmaf(w.x, Bk.x, fmaf(w.y, Bk.y,
                fmaf(w.z, Bk.z, fmaf(w.w, Bk.w, brow[k]))));
        }
        ex[i] = v;
        lmax  = fmaxf(lmax, v);
    }

    float m = waveMax(lmax);
    if (lane == 0) red[wave] = m;
    __syncthreads();
    float gmax = red[0];
    #pragma unroll
    for (int i = 1; i < 8; ++i) gmax = fmaxf(gmax, red[i]);
    __syncthreads();

    float lsum = 0.f;
    #pragma unroll
    for (int i = 0; i < P2_ITERS; ++i) {
        const int k = tid + i * P2_THREADS;
        ex[i] = (k < KPIX) ? __expf(ex[i] - gmax) : 0.f;
        lsum += ex[i];
    }
    float sm = waveSum(lsum);
    if (lane == 0) red[wave] = sm;
    __syncthreads();
    float gsum = 0.f;
    #pragma unroll
    for (int i = 0; i < 8; ++i) gsum += red[i];
    const float inv = __frcp_rn(gsum);

    float* __restrict__ outrow = (s < 4)
        ? zp + ((size_t)s * B_SZ + b) * KPIX
        : mean_profile + (size_t)b * KPIX;
    #pragma unroll
    for (int i = 0; i < P2_ITERS; ++i) {
        const int k = tid + i * P2_THREADS;
        if (k < KPIX) outrow[k] = ex[i] * inv;
    }
}

// ---------------------------------------------------------------------------
extern "C" void kernel_launch(void* const* d_in, const int* in_sizes, int n_in,
                              void* d_out, int out_size, void* d_ws, size_t ws_size,
                              hipStream_t stream) {
    const float* x     = (const float*)d_in[0];
    const float* eps   = (const float*)d_in[1];
    const float* A     = (const float*)d_in[2];
    const float* Bm    = (const float*)d_in[3];
    const float* C     = (const float*)d_in[4];
    const float* bias  = (const float*)d_in[5];
    const float* mu_w  = (const float*)d_in[6];
    const float* mu_b  = (const float*)d_in[7];
    const float* std_w = (const float*)d_in[8];
    const float* std_b = (const float*)d_in[9];
    const int*   gl    = (const int*)d_in[10];

    float* out     = (float*)d_out;
    float* zp      = out;
    float* meanp   = zp + (size_t)S_MC * B_SZ * KPIX;
    float* out_mu  = meanp + (size_t)B_SZ * KPIX;
    float* out_std = out_mu + (size_t)B_SZ * LAT;
    float* w_vecs  = (float*)d_ws;   // (B, 5, 4) floats = 2.62 MB

    phase1_encoder<<<B_SZ / 128, 256, 0, stream>>>(
        x, eps, A, C, mu_w, mu_b, std_w, std_b, gl, out_mu, out_std, w_vecs);

    phase2_softmax<<<(S_MC + 1) * B_SZ, P2_THREADS, 0, stream>>>(
        Bm, bias, gl, w_vecs, zp, meanp);
}